// GaussianRenderer_59330678227366
// MI455X (gfx1250) — compile-verified
//
#include <hip/hip_runtime.h>
#include <hip/hip_bf16.h>
#include <math.h>

// ---------------------------------------------------------------------------
// GaussianRenderer for MI455X (gfx1250, wave32, WMMA).
//
//   P = 128*128 pixels, N = 2048 gaussians, K = 20, THR = 0.01, TAU = 1.
//
// dot(pixel, gaussian) computed with V_WMMA_F32_16X16X4_F32 (K=3 padded to 4).
// Per-lane top-K kept as: g-values in registers (constant-indexed),
// idx/dot in LDS (dynamic-indexed) -> no scratch spills.
// ---------------------------------------------------------------------------

#define NG      2048
#define IMGW    128
#define KSEL    20
#define LOGTHR  (-4.605170186f)   // log(0.01)
#define FOCALI  (1.0f / 140.0f)
#define NEG_BIG (-3.0e38f)
#define POS_BIG ( 3.0e38f)
#define PI_F    3.14159265358979f

typedef float v2f __attribute__((ext_vector_type(2)));
typedef float v8f __attribute__((ext_vector_type(8)));

// ------------------------- phase 0: per-gaussian prep ----------------------
// gs layout (SoA): [0..N) qx | [N..2N) qy | [2N..3N) qz | [3N..4N) sigma
//                  | [4N..5N) sigma*|q|^2
__global__ void gr_prep_kernel(const float* __restrict__ verts,
                               const float* __restrict__ sigmas,
                               const float* __restrict__ R,
                               const float* __restrict__ T,
                               float* __restrict__ gs) {
    int n = blockIdx.x * blockDim.x + threadIdx.x;
    if (n >= NG) return;
    float vx = verts[3 * n + 0];
    float vy = verts[3 * n + 1];
    float vz = verts[3 * n + 2];
    // row-vector convention: c[k] = sum_j v[j] * R[j][k] + T[k]
    float cx = vx * R[0] + vy * R[3] + vz * R[6] + T[0];
    float cy = vx * R[1] + vy * R[4] + vz * R[7] + T[1];
    float cz = vx * R[2] + vy * R[5] + vz * R[8] + T[2];
    // q[i] = sum_j R[i][j] * c[j]
    float qx = R[0] * cx + R[1] * cy + R[2] * cz;
    float qy = R[3] * cx + R[4] * cy + R[5] * cz;
    float qz = R[6] * cx + R[7] * cy + R[8] * cz;
    float s  = sigmas[n];
    gs[n]          = qx;
    gs[NG + n]     = qy;
    gs[2 * NG + n] = qz;
    gs[3 * NG + n] = s;
    gs[4 * NG + n] = s * (qx * qx + qy * qy + qz * qz);
}

// ------------------------- phase 1: render ---------------------------------
// 1 wave handles 16 pixels. NWAVES waves per block.
#define NWAVES 4

__global__ void __launch_bounds__(32 * NWAVES)
gr_render_kernel(const float* __restrict__ gs,
                 const float* __restrict__ R,
                 float* __restrict__ outW,   // [P,20] weights
                 int*   __restrict__ outI,   // [P,20] indices (-1 invalid)
                 int*   __restrict__ outV,   // [P]    valid count
                 float* __restrict__ outH) { // [P,20] hit lengths
    // per pixel: 2*KSEL slots; half [0..20) owned by lane n16, half [20..40)
    // owned by lane n16+16. idx/dot are written here during the scan.
    __shared__ float sG[NWAVES][16][2 * KSEL];
    __shared__ int   sI[NWAVES][16][2 * KSEL];
    __shared__ float sD[NWAVES][16][2 * KSEL];

    const int lane = threadIdx.x & 31;
    const int wave = threadIdx.x >> 5;
    const int n16  = lane & 15;       // pixel column inside the wave tile
    const int hi   = lane >> 4;       // lane half: selects K{0,1} vs K{2,3}

    int*   myI = &sI[wave][n16][hi * KSEL];
    float* myD = &sD[wave][n16][hi * KSEL];
    for (int k = 0; k < KSEL; ++k) { myI[k] = -1; myD[k] = 0.0f; }

    // hint the gaussian table into cache early (global_prefetch_b8)
    __builtin_prefetch(gs + lane * 64, 0, 1);
    __builtin_prefetch(gs + 4 * NG + lane * 64, 0, 1);

    const int pix = (blockIdx.x * NWAVES + wave) * 16 + n16;
    const int x = pix & (IMGW - 1);
    const int y = pix >> 7;

    // --- unit ray for this pixel, rotated into gaussian frame ---
    float rx = (64.0f - (float)x) * FOCALI;
    float ry = (64.0f - (float)y) * FOCALI;
    float rz = 1.0f;
    float nrm = sqrtf(rx * rx + ry * ry + rz * rz);
    rx /= nrm; ry /= nrm; rz /= nrm;
    const float ax = R[0] * rx + R[1] * ry + R[2] * rz;
    const float ay = R[3] * rx + R[4] * ry + R[5] * rz;
    const float az = R[6] * rx + R[7] * ry + R[8] * rz;
    const float rn2 = ax * ax + ay * ay + az * az;
    const float inv_rn2 = 1.0f / rn2;

    // B matrix (4 x 16, rays as columns): VGPR0 = K0 (lanes 0-15) / K2
    // (lanes 16-31), VGPR1 = K1 / K3.  K3 row = 0 padding.
    v2f Bm;
    Bm.x = hi ? az : ax;
    Bm.y = hi ? 0.0f : ay;

    // per-lane top-K g-values in registers; current min cached.
    float topg[KSEL];
#pragma unroll
    for (int k = 0; k < KSEL; ++k) topg[k] = NEG_BIG;
    float curMin = NEG_BIG;
    int   curSlot = 0;

    for (int g0 = 0; g0 < NG; g0 += 16) {
        // A matrix (16 x 4, gaussians as rows): lanes 0-15 carry K0,K1,
        // lanes 16-31 carry K2,K3 (K3 = 0 padding).
        const int gi = g0 + n16;
        float qx = gs[gi];
        float qy = gs[NG + gi];
        float qz = gs[2 * NG + gi];
        v2f Am;
        Am.x = hi ? qz : qx;
        Am.y = hi ? 0.0f : qy;

        v8f D = {};
        D = __builtin_amdgcn_wmma_f32_16x16x4_f32(
                /*neg_a=*/false, Am, /*neg_b=*/false, Bm,
                /*c_mod=*/(short)0, D, /*reuse_a=*/false, /*reuse_b=*/false);

        // C/D layout: this lane holds pixel n16, gaussians m = r + 8*hi.
        const int mBase = g0 + hi * 8;
#pragma unroll
        for (int r = 0; r < 8; ++r) {
            const int   gidx = mBase + r;
            const float dot  = D[r];
            const float sig  = gs[3 * NG + gidx];
            const float cqq  = gs[4 * NG + gidx];
            const float gv   = sig * dot * dot * inv_rn2 - cqq; // log activation
            if (gv > LOGTHR && gv > curMin) {
                // replace current min slot (constant-indexed select)
#pragma unroll
                for (int j = 0; j < KSEL; ++j)
                    topg[j] = (j == curSlot) ? gv : topg[j];
                myI[curSlot] = gidx;     // dynamic index -> LDS, cheap
                myD[curSlot] = dot;
                // recompute min / argmin (unrolled, constant indices)
                float m = topg[0]; int ms = 0;
#pragma unroll
                for (int j = 1; j < KSEL; ++j)
                    if (topg[j] < m) { m = topg[j]; ms = j; }
                curMin = m; curSlot = ms;
            }
        }
    }

    // flush register g-values next to their idx/dot slots
#pragma unroll
    for (int k = 0; k < KSEL; ++k)
        sG[wave][n16][hi * KSEL + k] = topg[k];
    __syncthreads();

    if (hi == 0) {
        float* G  = sG[wave][n16];
        int*   I  = sI[wave][n16];
        float* Dd = sD[wave][n16];

        // top-KSEL by log-activation (descending) out of 2*KSEL candidates
        for (int k = 0; k < KSEL; ++k) {
            int best = k; float bg = G[k];
            for (int j = k + 1; j < 2 * KSEL; ++j)
                if (G[j] > bg) { bg = G[j]; best = j; }
            float tg = G[k]; G[k] = G[best]; G[best] = tg;
            float td = Dd[k]; Dd[k] = Dd[best]; Dd[best] = td;
            int   ti = I[k]; I[k] = I[best]; I[best] = ti;
        }

        // convert selected entries: G <- E, Dd <- hit length (invalid -> inf)
        for (int k = 0; k < KSEL; ++k) {
            float gk = G[k];
            if (gk > LOGTHR) {
                float sig = gs[3 * NG + I[k]];
                G[k]  = __expf(gk) * sqrtf(PI_F / (sig * rn2));
                Dd[k] = Dd[k] * inv_rn2;
            } else {
                G[k] = 0.0f; Dd[k] = POS_BIG; I[k] = -1;
            }
        }

        // sort the KSEL survivors by hit length ascending
        for (int k = 0; k < KSEL - 1; ++k) {
            int best = k; float bh = Dd[k];
            for (int j = k + 1; j < KSEL; ++j)
                if (Dd[j] < bh) { bh = Dd[j]; best = j; }
            float tg = G[k]; G[k] = G[best]; G[best] = tg;
            float td = Dd[k]; Dd[k] = Dd[best]; Dd[best] = td;
            int   ti = I[k]; I[k] = I[best]; I[best] = ti;
        }

        // exclusive-transmittance composite (TAU = 1) + outputs
        float cum = 0.0f;
        int nvalid = 0;
        for (int k = 0; k < KSEL; ++k) {
            float Ev = G[k];
            bool  valid = (I[k] >= 0);
            float Tk = __expf(-cum);
            outW[pix * KSEL + k] = valid ? (Ev * Tk) : 0.0f;
            outI[pix * KSEL + k] = I[k];
            outH[pix * KSEL + k] = valid ? Dd[k] : 0.0f;
            cum += Ev;
            nvalid += valid ? 1 : 0;
        }
        outV[pix] = nvalid;
    }
}

// ---------------------------------------------------------------------------
extern "C" void kernel_launch(void* const* d_in, const int* in_sizes, int n_in,
                              void* d_out, int out_size, void* d_ws, size_t ws_size,
                              hipStream_t stream) {
    const float* verts  = (const float*)d_in[0];  // [1,2048,3]
    const float* sigmas = (const float*)d_in[1];  // [2048]
    const float* R      = (const float*)d_in[2];  // [1,3,3]
    const float* T      = (const float*)d_in[3];  // [1,3]

    float* gs = (float*)d_ws;                     // 5*NG floats = 40 KB scratch

    // output tuple concatenated flat:
    //   vert_weight  [1,128,128,20] f32 : 327680
    //   vert_index   [1,128,128,20] i32 : 327680
    //   valid_num    [1,128,128]    i32 :  16384
    //   vert_hit_len [1,128,128,20] f32 : 327680
    float* out  = (float*)d_out;
    float* outW = out;
    int*   outI = (int*)(out + 327680);
    int*   outV = (int*)(out + 655360);
    float* outH = out + 671744;

    gr_prep_kernel<<<(NG + 255) / 256, 256, 0, stream>>>(verts, sigmas, R, T, gs);

    // 16384 pixels / (NWAVES waves * 16 px) = 256 blocks of 128 threads
    gr_render_kernel<<<256, 32 * NWAVES, 0, stream>>>(gs, R, outW, outI, outV, outH);
}